// DeformableConv2DLayer_84997402788306
// MI455X (gfx1250) — compile-verified
//
#include <hip/hip_runtime.h>

// fp32 WMMA fragments: A/B = 2 floats per lane, C/D = 8 floats per lane (wave32)
typedef __attribute__((ext_vector_type(2))) float v2f;
typedef __attribute__((ext_vector_type(8))) float v8f;

#define HH    96
#define WW    96
#define CIN   64
#define COUT  64
#define KKT   9            // 3x3 taps
#define KTOT  (CIN * KKT)  // 576 = GEMM K
#define HW    (HH * WW)    // 9216
#define TILE_PX 64
#define KCHUNK  16
// B tile stored k-pair-interleaved: elem (k,px) -> (k>>1)*PAIR_STRIDE + 2*px + (k&1)
// PAIR_STRIDE % 64 == 32 -> the two half-wave reads hit disjoint LDS bank halves.
#define PAIR_STRIDE 160

__launch_bounds__(256, 2)
__global__ void dcn_fused_wmma(const float* __restrict__ inp,
                               const float* __restrict__ off,
                               const float* __restrict__ msk,
                               const float* __restrict__ wgt,
                               float* __restrict__ out)
{
    __shared__ float4 s_w[KKT * TILE_PX];             // bilinear weights (mask+valid folded)
    __shared__ int4   s_i[KKT * TILE_PX];             // clamped corner indices (y*W+x)
    __shared__ float  s_b[(KCHUNK / 2) * PAIR_STRIDE];// staged im2col tile, pair-interleaved

    const int tid  = threadIdx.x;
    const int lane = tid & 31;
    const int wave = tid >> 5;

    const long tile = (long)blockIdx.x * TILE_PX;     // 9216 % 64 == 0 -> single image
    const int  n    = (int)(tile / HW);
    const int  hwb  = (int)(tile % HW);

    const float* inp_n = inp + (long)n * CIN * HW;
    const float* off_n = off + (long)n * (2 * KKT) * HW;
    const float* msk_n = msk + (long)n * KKT * HW;

    // ---------------- Phase A: per (tap, pixel) sampling metadata ----------------
    for (int e = tid; e < KKT * TILE_PX; e += 256) {
        int tap = e >> 6;              // 0..8
        int px  = e & 63;
        int hw  = hwb + px;
        int h   = hw / WW;
        int w   = hw - h * WW;
        int ky  = tap / 3;
        int kx  = tap - ky * 3;

        float oy = off_n[(tap * 2 + 0) * HW + hw];
        float ox = off_n[(tap * 2 + 1) * HW + hw];
        float mk = msk_n[tap * HW + hw];

        float py = oy + (float)(ky + h - 1);   // STRIDE=1, PAD=1, DIL=1
        float qx = ox + (float)(kx + w - 1);
        float y0f = floorf(py), x0f = floorf(qx);
        float wy1 = py - y0f,   wx1 = qx - x0f;
        float wy0 = 1.f - wy1,  wx0 = 1.f - wx1;
        int y0 = (int)y0f, x0 = (int)x0f;
        int y1 = y0 + 1,   x1 = x0 + 1;
        float vy0 = (y0 >= 0 && y0 < HH) ? 1.f : 0.f;
        float vy1 = (y1 >= 0 && y1 < HH) ? 1.f : 0.f;
        float vx0 = (x0 >= 0 && x0 < WW) ? 1.f : 0.f;
        float vx1 = (x1 >= 0 && x1 < WW) ? 1.f : 0.f;
        int cy0 = min(max(y0, 0), HH - 1), cy1 = min(max(y1, 0), HH - 1);
        int cx0 = min(max(x0, 0), WW - 1), cx1 = min(max(x1, 0), WW - 1);

        float4 wv;
        wv.x = mk * wy0 * wx0 * vy0 * vx0;
        wv.y = mk * wy0 * wx1 * vy0 * vx1;
        wv.z = mk * wy1 * wx0 * vy1 * vx0;
        wv.w = mk * wy1 * wx1 * vy1 * vx1;
        s_w[e] = wv;
        s_i[e] = make_int4(cy0 * WW + cx0, cy0 * WW + cx1,
                           cy1 * WW + cx0, cy1 * WW + cx1);
    }

    // wave -> (Cout block, pixel-block pair)
    const int mb  = wave >> 1;                 // 0..3  (16 Cout rows each)
    const int np  = wave & 1;                  // 0..1
    const int khalf = (lane >= 16) ? 2 : 0;    // fp32 WMMA half-wave K split
    const int l15   = lane & 15;
    const int rowA  = mb * 16 + l15;           // Cout row this lane feeds
    const int col0  = (np * 2 + 0) * 16 + l15; // pixel columns this lane feeds
    const int col1  = (np * 2 + 1) * 16 + l15;

    // kc-invariant fill-thread coordinates and B-store offsets
    int kl_c[4], px_c[4], soff_c[4];
    #pragma unroll
    for (int i = 0; i < 4; ++i) {
        int e = tid + i * 256;
        kl_c[i]   = e >> 6;                    // 0..15
        px_c[i]   = e & 63;
        soff_c[i] = (kl_c[i] >> 1) * PAIR_STRIDE + 2 * px_c[i] + (kl_c[i] & 1);
    }

    v8f acc0 = {};
    v8f acc1 = {};

    float g[16];   // pipelined gather registers: 4 B-elements x 4 corners
    int   rr[4];   // tap-table record per element

    __syncthreads();   // s_w / s_i ready

    // prologue: issue gathers for chunk 0
    #pragma unroll
    for (int i = 0; i < 4; ++i) {
        int kk  = kl_c[i];                     // kc = 0
        int c   = kk / KKT;
        int tap = kk - c * KKT;
        rr[i]   = tap * TILE_PX + px_c[i];
        int4 iv = s_i[rr[i]];
        const float* bc = inp_n + c * HW;
        g[i * 4 + 0] = bc[iv.x];
        g[i * 4 + 1] = bc[iv.y];
        g[i * 4 + 2] = bc[iv.z];
        g[i * 4 + 3] = bc[iv.w];
    }

    // ---------------- Main loop over GEMM K = (c, tap) ----------------
    for (int kc = 0; kc < KTOT; kc += KCHUNK) {
        // reduce + store current chunk into s_b
        #pragma unroll
        for (int i = 0; i < 4; ++i) {
            float4 wv = s_w[rr[i]];
            s_b[soff_c[i]] = wv.x * g[i * 4 + 0] + wv.y * g[i * 4 + 1]
                           + wv.z * g[i * 4 + 2] + wv.w * g[i * 4 + 3];
        }
        __syncthreads();

        // issue gathers for the NEXT chunk; overlaps the WMMA consume below
        if (kc + KCHUNK < KTOT) {
            #pragma unroll
            for (int i = 0; i < 4; ++i) {
                int kk  = kc + KCHUNK + kl_c[i];
                int c   = kk / KKT;
                int tap = kk - c * KKT;
                rr[i]   = tap * TILE_PX + px_c[i];
                int4 iv = s_i[rr[i]];
                const float* bc = inp_n + c * HW;
                g[i * 4 + 0] = bc[iv.x];
                g[i * 4 + 1] = bc[iv.y];
                g[i * 4 + 2] = bc[iv.z];
                g[i * 4 + 3] = bc[iv.w];
            }
        }

        // consume: 4 WMMA k-steps on the chunk just staged
        #pragma unroll
        for (int s = 0; s < 4; ++s) {
            int ks   = s * 4 + khalf;          // even -> b64-aligned pair
            int pbase = (ks >> 1) * PAIR_STRIDE;
            v2f a  = *(const v2f*)(wgt + (long)rowA * KTOT + kc + ks);
            v2f b0 = *(const v2f*)(s_b + pbase + 2 * col0);
            v2f b1 = *(const v2f*)(s_b + pbase + 2 * col1);
            acc0 = __builtin_amdgcn_wmma_f32_16x16x4_f32(
                       false, a, false, b0, (short)0, acc0, false, false);
            acc1 = __builtin_amdgcn_wmma_f32_16x16x4_f32(
                       false, a, false, b1, (short)0, acc1, false, false);
        }
        __syncthreads();
    }

    // ---------------- Store C/D tiles ----------------
    const int mrow = (lane >= 16) ? 8 : 0;     // C/D layout: high lanes hold M+8
    #pragma unroll
    for (int r = 0; r < 8; ++r) {
        int  o    = mb * 16 + mrow + r;
        long base = ((long)n * COUT + o) * HW + hwb;
        out[base + col0] = acc0[r];
        out[base + col1] = acc1[r];
    }
}

extern "C" void kernel_launch(void* const* d_in, const int* in_sizes, int n_in,
                              void* d_out, int out_size, void* d_ws, size_t ws_size,
                              hipStream_t stream) {
    const float* inp = (const float*)d_in[0];   // [8,64,96,96]
    const float* off = (const float*)d_in[1];   // [8,18,96,96]
    const float* msk = (const float*)d_in[2];   // [8,9,96,96]
    const float* wgt = (const float*)d_in[3];   // [64,64,3,3]
    float* out = (float*)d_out;                 // [8,64,96,96]

    const int total_px = 8 * HW;                // 73728
    const int nblocks  = total_px / TILE_PX;    // 1152
    dcn_fused_wmma<<<nblocks, 256, 0, stream>>>(inp, off, msk, wgt, out);
}